// CRF_78537771974733
// MI455X (gfx1250) — compile-verified
//
#include <hip/hip_runtime.h>

#define NPART 1024
#define BLOCK 256

typedef __attribute__((ext_vector_type(2))) float v2f;
typedef __attribute__((ext_vector_type(8))) float v8f;

// ---------------------------------------------------------------------------
// Kernel 1: gather + per-block deterministic reduction.
// Emits: global loads (gather), ds loads/stores (LDS transitions),
//        global_prefetch_b8 (CDNA5 prefetch path).
// ---------------------------------------------------------------------------
__global__ void crf_partial_kernel(const float* __restrict__ logits,
                                   const int*   __restrict__ tags,
                                   const float* __restrict__ trans,
                                   float*       __restrict__ partials,
                                   int S, int T) {
    __shared__ float trans_s[2560];            // T*T <= 2500 for T=50
    const int tid = threadIdx.x;
    const int nTT = T * T;
    for (int j = tid; j < nTT; j += BLOCK) trans_s[j] = trans[j];
    __syncthreads();

    const long long stride = (long long)gridDim.x * BLOCK;
    float acc = 0.0f;
    for (long long i = (long long)blockIdx.x * BLOCK + tid; i < S; i += stride) {
        // speculative prefetch of next-stride data (safe if OOB: dropped)
        __builtin_prefetch(tags + (i + stride), 0, 1);
        __builtin_prefetch(logits + (size_t)(i + stride) * (size_t)T, 0, 1);

        const int t = tags[i];
        acc += logits[(size_t)i * (size_t)T + (size_t)t];
        if (i > 0) {
            const int tp = tags[i - 1];
            acc += trans_s[tp * T + t];
        }
    }

    // wave32 butterfly reduction (deterministic order)
    #pragma unroll
    for (int off = 16; off > 0; off >>= 1)
        acc += __shfl_down(acc, off, 32);

    __shared__ float wsum[BLOCK / 32];
    const int wave = tid >> 5;
    if ((tid & 31) == 0) wsum[wave] = acc;
    __syncthreads();

    if (tid == 0) {
        float b = 0.0f;
        #pragma unroll
        for (int w = 0; w < BLOCK / 32; ++w) b += wsum[w];
        partials[blockIdx.x] = b;
    }
}

// ---------------------------------------------------------------------------
// Kernel 2: final reduction of NPART partials on the matrix pipe.
// One wave; 16 chained V_WMMA_F32_16X16X4_F32 with B = ones:
//   D[m][n] = sum over chunks of sum_k A[m][k]   (same value in every column)
// Column 0 of D lives in lane 0 (M=0..7) and lane 16 (M=8..15).
// ---------------------------------------------------------------------------
__global__ void crf_final_kernel(const float* __restrict__ partials,
                                 float*       __restrict__ out) {
    const int lane = threadIdx.x;          // 0..31, single wave, EXEC all ones
    const int m  = lane & 15;              // A-matrix row
    const int kb = (lane >> 4) << 1;       // K base: lanes 0-15 -> 0, 16-31 -> 2

    v2f bmat; bmat.x = 1.0f; bmat.y = 1.0f;   // B = all ones (layout-agnostic)
    v8f d = {};

    #pragma unroll
    for (int c = 0; c < NPART / 64; ++c) {
        const int base = c * 64;
        v2f a;
        a.x = partials[base + m * 4 + kb + 0];   // A[m][kb]
        a.y = partials[base + m * 4 + kb + 1];   // A[m][kb+1]
        // 8 args: (neg_a, A, neg_b, B, c_mod, C, reuse_a, reuse_b)
        d = __builtin_amdgcn_wmma_f32_16x16x4_f32(false, a, false, bmat,
                                                  (short)0, d, false, false);
    }

    float s = d[0] + d[1] + d[2] + d[3] + d[4] + d[5] + d[6] + d[7];
    const float hi = __shfl_down(s, 16, 32);   // lane16's M=8..15 sums (N=0)
    if (lane == 0) out[0] = -(s + hi);
}

// ---------------------------------------------------------------------------
// Launch
// ---------------------------------------------------------------------------
extern "C" void kernel_launch(void* const* d_in, const int* in_sizes, int n_in,
                              void* d_out, int out_size, void* d_ws, size_t ws_size,
                              hipStream_t stream) {
    const float* logits = (const float*)d_in[0];
    const int*   tags   = (const int*)d_in[1];
    const float* trans  = (const float*)d_in[2];

    const int S = in_sizes[1];               // 2,000,000
    const int T = in_sizes[0] / S;           // 50

    float* partials = (float*)d_ws;          // NPART floats of scratch

    crf_partial_kernel<<<NPART, BLOCK, 0, stream>>>(logits, tags, trans,
                                                    partials, S, T);
    crf_final_kernel<<<1, 32, 0, stream>>>(partials, (float*)d_out);
}